// TorchColorCorrectorWavelet_13477607375402
// MI455X (gfx1250) — compile-verified
//
#include <hip/hip_runtime.h>

// Problem constants (fixed by the reference: shape (1,3,16,720,1280) f32)
#define IMG_H   720
#define IMG_W   1280
#define PLANES  48          // 3 channels * 16 frames; layout identical per plane
#define TILE    64
#define HALO    31          // cumulative radius of 5 dilated blurs: 1+2+4+8+16
#define REG     128         // padded region extent (>= 64 + 2*31 = 126), multiple of 4
#define LD_V    132         // V / T row stride (mod 64 == 4 -> conflict-free A-frags)
#define PRD     288         // D pair-row stride  (2*144; mod 64 == 32 -> lane halves disjoint)
#define PRH     160         // H pair-row stride  (2*80;  mod 64 == 32)
#define BAND_W  64          // band table width: offsets -31..+31 (slot 63 always 0)

typedef __attribute__((ext_vector_type(2))) float v2f;
typedef __attribute__((ext_vector_type(8))) float v8f;

// ---------------------------------------------------------------------------
// Setup: compose the 5 clamped banded blur matrices into a 63-tap band per row.
// Row r of V = e_r^T * Mv16*Mv8*Mv4*Mv2*Mv1 (levels applied to X in order 1..16,
// so row extraction multiplies by Mv16 first). All taps provably stay within
// offset [-31,+31] of r; mass clamped outside the image folds onto rows 0/N-1.
// band[r][d] = weight for source row (r + d - 31); entries for out-of-image
// sources remain exactly 0.
// ---------------------------------------------------------------------------
__global__ void build_bands_kernel(float* __restrict__ bandV,
                                   float* __restrict__ bandH) {
  int t = blockIdx.x * blockDim.x + threadIdx.x;
  int N, r;
  float* outp;
  if (t < IMG_H)              { N = IMG_H; r = t;          outp = bandV + t * BAND_W; }
  else if (t < IMG_H + IMG_W) { N = IMG_W; r = t - IMG_H;  outp = bandH + (t - IMG_H) * BAND_W; }
  else return;

  float cur[BAND_W], nxt[BAND_W];
  for (int i = 0; i < BAND_W; ++i) cur[i] = 0.f;
  cur[HALO] = 1.f;  // delta at offset 0
  for (int lvl = 4; lvl >= 0; --lvl) {   // radii 16,8,4,2,1
    int R = 1 << lvl;
    for (int i = 0; i < BAND_W; ++i) nxt[i] = 0.f;
    for (int d = 0; d < BAND_W; ++d) {
      float w = cur[d];
      if (w != 0.f) {
        int g  = r + d - HALO;           // in-image by construction
        int gm = g - R; if (gm < 0)     gm = 0;
        int gp = g + R; if (gp > N - 1) gp = N - 1;
        nxt[gm - r + HALO] += 0.25f * w;
        nxt[d]             += 0.50f * w;
        nxt[gp - r + HALO] += 0.25f * w;
      }
    }
    for (int i = 0; i < BAND_W; ++i) cur[i] = nxt[i];
  }
  for (int i = 0; i < BAND_W; ++i) outp[i] = cur[i];
}

// ---------------------------------------------------------------------------
// Main kernel: per 64x64 tile, out = clip(hq - V_blk * (hq-lq) * H_blk^T).
//   GEMM1: T(64x128)  = V_blk(64x128) * D(128x128)      (K = vertical region)
//   GEMM2: O(64x64)   = T(64x128)     * H_blk^T(128x64) (K = horizontal region)
// Both GEMMs run entirely on V_WMMA_F32_16X16X4_F32 with zero-filled K tails.
// B operands (D, H) use a K-pair-interleaved LDS layout so each lane's
// B-fragment {K, K+1} is a single contiguous ds_read_b64 (no packing movs).
// ---------------------------------------------------------------------------
__global__ __launch_bounds__(256) void wavelet_cc_kernel(
    const float* __restrict__ hq, const float* __restrict__ lq,
    const float* __restrict__ bandV, const float* __restrict__ bandH,
    float* __restrict__ out) {
  extern __shared__ float lds[];
  float* Dp = lds;                   // 64 pair-rows x PRD   D = hq-lq, (k,c) -> [k>>1][2c + (k&1)]
  float* Vd = Dp + 64 * PRD;         // TILE x LD_V          V block, row-major [m][k]
  float* Td = Vd + TILE * LD_V;      // TILE x LD_V          T = V*D, row-major [m][k]
  float* Hp = Td + TILE * LD_V;      // 64 pair-rows x PRH   H block, (k,j) -> [k>>1][2j + (k&1)]

  const int tid = threadIdx.x;
  const int tx  = blockIdx.x * TILE;
  const int ty  = blockIdx.y * TILE;
  const int rv0 = max(0, ty - HALO);
  const int hR  = min(IMG_H, ty + TILE + HALO) - rv0;   // <= 126
  const int cx0 = max(0, tx - HALO);
  const int wR  = min(IMG_W, tx + TILE + HALO) - cx0;   // <= 126
  const int pbase = blockIdx.z * (IMG_H * IMG_W);

  // ---- Phase A: stage V (row-major) and H (K-pair-interleaved) from band tables
  for (int idx = tid; idx < TILE * REG; idx += 256) {
    int m = idx >> 7, k = idx & (REG - 1);
    int r = ty + m;
    int d = (rv0 + k) - r + HALO;
    float v = 0.f;
    if (r < IMG_H && (unsigned)d < 64u) v = bandV[r * BAND_W + d];
    Vd[m * LD_V + k] = v;
  }
  for (int idx = tid; idx < REG * TILE; idx += 256) {
    int k = idx >> 6, j = idx & 63;
    int c = tx + j;
    int d = (cx0 + k) - c + HALO;
    float v = 0.f;
    if (c < IMG_W && (unsigned)d < 64u) v = bandH[c * BAND_W + d];
    Hp[(k >> 1) * PRH + 2 * j + (k & 1)] = v;
  }
  // ---- Phase B: D = hq - lq, stored K-pair-interleaved; consecutive threads
  // write consecutive LDS addresses (conflict-free stores).
  for (int idx = tid; idx < REG * REG; idx += 256) {
    int t   = idx >> 8;          // pair-row (k = 2t + par)
    int rem = idx & 255;         // rem = 2*c + par
    int c   = rem >> 1;
    int k   = 2 * t + (rem & 1);
    float v = 0.f;
    if (k < hR && c < wR) {
      int off = pbase + (rv0 + k) * IMG_W + (cx0 + c);
      v = hq[off] - lq[off];
    }
    Dp[t * PRD + rem] = v;
  }
  __syncthreads();

  const int lane = tid & 31;
  const int wave = tid >> 5;
  const int lm = lane & 15;   // A: M index / B: N index
  const int lh = lane >> 4;   // selects K pair {0,1} vs {2,3}

  // ---- Phase C: GEMM1  T = V * D   (8 waves, each a 32x32 supertile = 2x2 WMMA blocks)
  {
    const int bmp = wave >> 2;                 // 0..1
    const int bnp = wave & 3;                  // 0..3
    const float* a0p = Vd + (bmp * 32 + lm) * LD_V + 2 * lh;
    const float* a1p = a0p + 16 * LD_V;
    const float* bp  = Dp + lh * PRD + (bnp * 32 + lm) * 2;
    v8f acc00 = {}, acc01 = {}, acc10 = {}, acc11 = {};
    for (int k0 = 0; k0 < REG; k0 += 4) {
      v2f a0 = *(const v2f*)(a0p + k0);                    // ds_read_b64
      v2f a1 = *(const v2f*)(a1p + k0);
      const float* bk = bp + (k0 >> 1) * PRD;
      v2f b0 = *(const v2f*)(bk);                          // {D[k][c], D[k+1][c]}
      v2f b1 = *(const v2f*)(bk + 32);                     // col block +16
      acc00 = __builtin_amdgcn_wmma_f32_16x16x4_f32(false, a0, false, b0, (short)0, acc00, false, false);
      acc01 = __builtin_amdgcn_wmma_f32_16x16x4_f32(false, a0, false, b1, (short)0, acc01, false, false);
      acc10 = __builtin_amdgcn_wmma_f32_16x16x4_f32(false, a1, false, b0, (short)0, acc10, false, false);
      acc11 = __builtin_amdgcn_wmma_f32_16x16x4_f32(false, a1, false, b1, (short)0, acc11, false, false);
    }
    // C/D layout: VGPR v -> row = 8*lh + v, col = lm (within 16x16 block)
    const int mr = bmp * 32 + 8 * lh;
    const int nc = bnp * 32 + lm;
    for (int v = 0; v < 8; ++v) {
      Td[(mr + v) * LD_V + nc]           = acc00[v];
      Td[(mr + v) * LD_V + nc + 16]      = acc01[v];
      Td[(mr + 16 + v) * LD_V + nc]      = acc10[v];
      Td[(mr + 16 + v) * LD_V + nc + 16] = acc11[v];
    }
  }
  __syncthreads();

  // ---- Phase D: GEMM2  O = T * H^T  (8 waves, each 1x2 WMMA blocks) + epilogue
  {
    const int bm  = wave >> 1;                 // 0..3
    const int bn0 = (wave & 1) * 2;            // 0 or 2
    const float* ap = Td + (bm * 16 + lm) * LD_V + 2 * lh;
    const float* bp = Hp + lh * PRH + (bn0 * 16 + lm) * 2;
    v8f acc0 = {}, acc1 = {};
    for (int k0 = 0; k0 < REG; k0 += 4) {
      v2f a = *(const v2f*)(ap + k0);
      const float* bk = bp + (k0 >> 1) * PRH;
      v2f b0 = *(const v2f*)(bk);
      v2f b1 = *(const v2f*)(bk + 32);
      acc0 = __builtin_amdgcn_wmma_f32_16x16x4_f32(false, a, false, b0, (short)0, acc0, false, false);
      acc1 = __builtin_amdgcn_wmma_f32_16x16x4_f32(false, a, false, b1, (short)0, acc1, false, false);
    }
    // out = clip(hq - V*D*H^T).  x is always in-bounds (1280 % 64 == 0); guard y.
    const int gy0 = ty + bm * 16 + 8 * lh;
    const int gx0 = tx + bn0 * 16 + lm;
    for (int v = 0; v < 8; ++v) {
      int gy = gy0 + v;
      if (gy < IMG_H) {
        int off = pbase + gy * IMG_W;
        float v0 = hq[off + gx0] - acc0[v];
        float v1 = hq[off + gx0 + 16] - acc1[v];
        out[off + gx0]      = fminf(1.f, fmaxf(-1.f, v0));
        out[off + gx0 + 16] = fminf(1.f, fmaxf(-1.f, v1));
      }
    }
  }
}

// ---------------------------------------------------------------------------
extern "C" void kernel_launch(void* const* d_in, const int* in_sizes, int n_in,
                              void* d_out, int out_size, void* d_ws, size_t ws_size,
                              hipStream_t stream) {
  (void)in_sizes; (void)n_in; (void)out_size; (void)ws_size;
  const float* hq = (const float*)d_in[0];
  const float* lq = (const float*)d_in[1];
  float* outp = (float*)d_out;

  // Workspace: band tables (720*64 + 1280*64 floats = 512 KB), rebuilt every call.
  float* bandV = (float*)d_ws;
  float* bandH = bandV + IMG_H * BAND_W;

  build_bands_kernel<<<(IMG_H + IMG_W + 127) / 128, 128, 0, stream>>>(bandV, bandH);

  const size_t shmem =
      (size_t)(64 * PRD + 2 * TILE * LD_V + 64 * PRH) * sizeof(float); // ~178 KB
  (void)hipFuncSetAttribute((const void*)wavelet_cc_kernel,
                            hipFuncAttributeMaxDynamicSharedMemorySize, (int)shmem);

  dim3 grid(IMG_W / TILE, (IMG_H + TILE - 1) / TILE, PLANES);  // 20 x 12 x 48
  wavelet_cc_kernel<<<grid, 256, shmem, stream>>>(hq, lq, bandV, bandH, outp);
}